// PPNBaseline_74251394613728
// MI455X (gfx1250) — compile-verified
//
#include <hip/hip_runtime.h>
#include <math.h>

// ---------------------------------------------------------------------------
// Problem constants (fixed by the reference)
// ---------------------------------------------------------------------------
#define N_NODES 50000
#define N_EDGES 800000
#define D_HID   128     // H1 == H2 == 128
#define K_MAX   256     // largest GEMM K (D_IN and H1+H2)
#define LDS_PAD 4       // row-stride pad (floats): keeps 16B align, kills bank conflicts

typedef __attribute__((ext_vector_type(2))) float v2f;
typedef __attribute__((ext_vector_type(8))) float v8f;
typedef __attribute__((ext_vector_type(4))) float f32x4;

// b128 payload type expected by the async-LDS builtin (GCC vector style, per
// the compiler diagnostic: "__vector_size__(4 * sizeof(int)) int").
// Param 1 is a NON-const AS1 pointer (diagnostic round 3), param 2 is AS3.
typedef int vi4 __attribute__((vector_size(16)));
typedef __attribute__((address_space(1))) vi4* gvi4_ptr;
typedef __attribute__((address_space(3))) vi4* lvi4_ptr;

// ---------------------------------------------------------------------------
// gfx1250 async global->LDS copy helpers (ASYNCcnt-tracked), with fallbacks.
// ---------------------------------------------------------------------------
#if defined(__has_builtin)
#if __has_builtin(__builtin_amdgcn_global_load_async_to_lds_b128)
#define HAVE_ASYNC_LDS 1
#endif
#endif

__device__ __forceinline__ void g2lds_b128(const float* g, float* l) {
#ifdef HAVE_ASYNC_LDS
  // generic->AS1: same 64-bit value; generic->AS3: low 32 bits are the LDS
  // offset (addrspacecast AS3->flat is {aperture_hi, offset32}).
  gvi4_ptr gp = (gvi4_ptr)(uintptr_t)(const void*)g;
  lvi4_ptr lp = (lvi4_ptr)(unsigned)(uintptr_t)(void*)l;
  __builtin_amdgcn_global_load_async_to_lds_b128(gp, lp, 0, 0);
#else
  *(f32x4*)l = *(const f32x4*)g;
#endif
}

__device__ __forceinline__ void async_wait0() {
#ifdef HAVE_ASYNC_LDS
#if __has_builtin(__builtin_amdgcn_s_wait_asynccnt)
  __builtin_amdgcn_s_wait_asynccnt(0);
#else
  asm volatile("s_wait_asynccnt 0" ::: "memory");
#endif
#endif
}

// ---------------------------------------------------------------------------
// Ordered-uint encoding of floats: monotone bijection so unsigned atomicMax
// implements float segment-max (handles negatives / -inf correctly).
// ---------------------------------------------------------------------------
__device__ __forceinline__ unsigned f2ord(float f) {
  unsigned u = __float_as_uint(f);
  return (u & 0x80000000u) ? ~u : (u | 0x80000000u);
}
__device__ __forceinline__ float ord2f(unsigned u) {
  unsigned b = (u & 0x80000000u) ? (u ^ 0x80000000u) : ~u;
  return __uint_as_float(b);
}

// ---------------------------------------------------------------------------
// WMMA fp32 GEMM:  Y[M x NC] = X[M x K](ld=ldx) * W[NC x K]^T + bias
//   grid.x  = M / 16
//   blockDim.x = 32 * (NC / 16); wave w computes column tile w.
// The 16xK X panel for this block is staged once into LDS via async
// global->LDS B128 copies (all waves then read A fragments with ds_load_b64),
// removing the 8x-redundant global reads of X across waves.
// Fragment layouts per ISA 7.12.2 (fp32 16x16x4):
//   A 16x4 : lanes 0-15 -> M=lane, VGPR j holds K=j (K=j+2 for lanes 16-31)
//   B 4x16 : lanes 0-15 -> N=lane, VGPR j holds K=j (K=j+2 for lanes 16-31)
//   C 16x16: VGPR i -> M=i (lanes 0-15) / M=i+8 (lanes 16-31), N=lane&15
// ---------------------------------------------------------------------------
__global__ void gemm_wmma_f32(const float* __restrict__ X, int ldx,
                              const float* __restrict__ W,
                              const float* __restrict__ bias,
                              float* __restrict__ Y, int ldy,
                              int K) {
  __shared__ float sX[16 * (K_MAX + LDS_PAD)];
  const int lstride = K + LDS_PAD;       // padded LDS row stride (floats)

  // ---- Stage X[16 rows x K] panel into LDS (async B128 chunks) ----
  const int kq = K >> 2;                 // 4-float chunks per row
  const int mbase = blockIdx.x << 4;
  for (int c = threadIdx.x; c < (K << 2); c += blockDim.x) {  // 16*K/4 chunks
    const int row  = c / kq;
    const int kpos = (c - row * kq) << 2;
    g2lds_b128(X + (size_t)(mbase + row) * (size_t)ldx + kpos,
               &sX[row * lstride + kpos]);
  }
  async_wait0();
  __syncthreads();

  // ---- WMMA main loop ----
  const int lane = threadIdx.x & 31;
  const int wave = threadIdx.x >> 5;
  const int arow = lane & 15;
  const int ncol = (wave << 4) + (lane & 15);
  const int koff = (lane >> 4) << 1;     // 0 for lanes 0-15, 2 for lanes 16-31

  const float* ap = &sX[arow * lstride + koff];
  const float* wp = W + (size_t)ncol * (size_t)K + koff;

  v8f acc = {0.f, 0.f, 0.f, 0.f, 0.f, 0.f, 0.f, 0.f};
  for (int kk = 0; kk < K; kk += 4) {
    v2f a, b;
    a.x = ap[kk];     a.y = ap[kk + 1];
    b.x = wp[kk];     b.y = wp[kk + 1];
    acc = __builtin_amdgcn_wmma_f32_16x16x4_f32(
        /*neg_a=*/false, a, /*neg_b=*/false, b,
        /*c_mod=*/(short)0, acc, /*reuse_a=*/false, /*reuse_b=*/false);
  }

  const int rbase = mbase + ((lane >> 4) << 3);  // +0 or +8
  const int c     = (wave << 4) + (lane & 15);
  const float bv  = bias ? bias[c] : 0.0f;
#pragma unroll
  for (int i = 0; i < 8; ++i) {
    Y[(size_t)(rbase + i) * (size_t)ldy + c] = acc[i] + bv;
  }
}

// ---------------------------------------------------------------------------
// Attention helper kernels
// ---------------------------------------------------------------------------

// Zero aggregate buffer, init per-node running max (= -inf encoded) and denom.
__global__ void init_attn(float* __restrict__ agg,
                          unsigned* __restrict__ mord,
                          float* __restrict__ denom) {
  const int i = blockIdx.x * blockDim.x + threadIdx.x;
  if (i < N_NODES * D_HID) agg[i] = 0.0f;
  if (i < N_NODES) {
    mord[i]  = f2ord(-INFINITY);
    denom[i] = 0.0f;
  }
}

// One wave per edge: score = dot(q[dst], k[src] + e) * inv_sqrt_d;
// segment-max into mord[dst].
__global__ void edge_score(const int* __restrict__ src, const int* __restrict__ dst,
                           const float* __restrict__ q, const float* __restrict__ k,
                           const float* __restrict__ ebuf,
                           float* __restrict__ score, unsigned* __restrict__ mord,
                           float inv_sqrt_d) {
  const int e = blockIdx.x * (blockDim.x >> 5) + (threadIdx.x >> 5);
  const int lane = threadIdx.x & 31;
  if (e >= N_EDGES) return;
  const int s = src[e], d = dst[e];
  const float* qp = q + (size_t)d * D_HID;
  const float* kp = k + (size_t)s * D_HID;
  const float* ep = ebuf + (size_t)e * D_HID;
  float p = 0.0f;
#pragma unroll
  for (int j = 0; j < 4; ++j) {
    const int c = lane + 32 * j;
    p += qp[c] * (kp[c] + ep[c]);
  }
#pragma unroll
  for (int off = 16; off > 0; off >>= 1) p += __shfl_down(p, off, 32);
  if (lane == 0) {
    const float sc = p * inv_sqrt_d;
    score[e] = sc;
    atomicMax(&mord[d], f2ord(sc));
  }
}

// Per node: decode max; nodes with no incoming edges (-inf) -> 0 (ref semantics).
__global__ void finalize_max(const unsigned* __restrict__ mord,
                             float* __restrict__ m) {
  const int i = blockIdx.x * blockDim.x + threadIdx.x;
  if (i >= N_NODES) return;
  const float s = ord2f(mord[i]);
  m[i] = isfinite(s) ? s : 0.0f;
}

// Per edge: ex = exp(score - m[dst]); accumulate denom[dst]; overwrite score.
__global__ void edge_exp(const int* __restrict__ dst,
                         float* __restrict__ score,
                         const float* __restrict__ m,
                         float* __restrict__ denom) {
  const int e = blockIdx.x * blockDim.x + threadIdx.x;
  if (e >= N_EDGES) return;
  const int d = dst[e];
  const float ex = expf(score[e] - m[d]);
  score[e] = ex;
  atomicAdd(&denom[d], ex);
}

// One wave per edge: agg[dst] += alpha * (v[src] + e), alpha = ex / (denom+eps).
__global__ void edge_aggregate(const int* __restrict__ src, const int* __restrict__ dst,
                               const float* __restrict__ score,
                               const float* __restrict__ denom,
                               const float* __restrict__ v,
                               const float* __restrict__ ebuf,
                               float* __restrict__ agg) {
  const int e = blockIdx.x * (blockDim.x >> 5) + (threadIdx.x >> 5);
  const int lane = threadIdx.x & 31;
  if (e >= N_EDGES) return;
  const int s = src[e], d = dst[e];
  const float alpha = score[e] / (denom[d] + 1e-16f);
  const float* vp = v + (size_t)s * D_HID;
  const float* ep = ebuf + (size_t)e * D_HID;
  float* op = agg + (size_t)d * D_HID;
#pragma unroll
  for (int j = 0; j < 4; ++j) {
    const int c = lane + 32 * j;
    atomicAdd(&op[c], alpha * (vp[c] + ep[c]));
  }
}

// One wave per node: beta = sigmoid(Wb . [agg, xr, agg-xr]);
// y = leaky_relu(beta*xr + (1-beta)*agg) written into xcat[:, colOff:colOff+128].
__global__ void beta_gate(const float* __restrict__ agg,
                          const float* __restrict__ xr,
                          const float* __restrict__ Wb,
                          float* __restrict__ xcat, int colOff) {
  const int n = blockIdx.x * (blockDim.x >> 5) + (threadIdx.x >> 5);
  const int lane = threadIdx.x & 31;
  if (n >= N_NODES) return;
  float ov[4], xv[4];
  float p = 0.0f;
#pragma unroll
  for (int j = 0; j < 4; ++j) {
    const int c = lane + 32 * j;
    ov[j] = agg[(size_t)n * D_HID + c];
    xv[j] = xr[(size_t)n * D_HID + c];
    p += Wb[c] * ov[j] + Wb[D_HID + c] * xv[j] + Wb[2 * D_HID + c] * (ov[j] - xv[j]);
  }
#pragma unroll
  for (int off = 16; off > 0; off >>= 1) p += __shfl_down(p, off, 32);
  p = __shfl(p, 0, 32);
  const float beta = 1.0f / (1.0f + expf(-p));
#pragma unroll
  for (int j = 0; j < 4; ++j) {
    const int c = lane + 32 * j;
    float y = beta * xv[j] + (1.0f - beta) * ov[j];
    y = (y >= 0.0f) ? y : 0.01f * y;  // LeakyReLU(0.01)
    xcat[(size_t)n * 256 + colOff + c] = y;
  }
}

// ---------------------------------------------------------------------------
// Host launcher
// ---------------------------------------------------------------------------
extern "C" void kernel_launch(void* const* d_in, const int* in_sizes, int n_in,
                              void* d_out, int out_size, void* d_ws, size_t ws_size,
                              hipStream_t stream) {
  (void)in_sizes; (void)n_in; (void)out_size; (void)ws_size;

  const float* x_in   = (const float*)d_in[0];
  const int*   eidx   = (const int*)d_in[1];
  const float* eatt   = (const float*)d_in[2];
  const float* Wq1    = (const float*)d_in[3];
  const float* bq1    = (const float*)d_in[4];
  const float* Wk1    = (const float*)d_in[5];
  const float* bk1    = (const float*)d_in[6];
  const float* Wv1    = (const float*)d_in[7];
  const float* bv1    = (const float*)d_in[8];
  const float* We1    = (const float*)d_in[9];
  const float* Wskip1 = (const float*)d_in[10];
  const float* bskip1 = (const float*)d_in[11];
  const float* Wbeta1 = (const float*)d_in[12];
  const float* Wq2    = (const float*)d_in[13];
  const float* bq2    = (const float*)d_in[14];
  const float* Wk2    = (const float*)d_in[15];
  const float* bk2    = (const float*)d_in[16];
  const float* Wv2    = (const float*)d_in[17];
  const float* bv2    = (const float*)d_in[18];
  const float* We2    = (const float*)d_in[19];
  const float* Wskip2 = (const float*)d_in[20];
  const float* bskip2 = (const float*)d_in[21];
  const float* Wbeta2 = (const float*)d_in[22];
  const float* Wm     = (const float*)d_in[23];
  const float* bm     = (const float*)d_in[24];
  float* outp = (float*)d_out;

  const int* src = eidx;            // edge_index[0]
  const int* dst = eidx + N_EDGES;  // edge_index[1]

  // Workspace carve-up (floats). Total ~592 MB.
  float* ws = (float*)d_ws;
  size_t off = 0;
  auto alloc = [&](size_t n) { float* p = ws + off; off += n; return p; };
  float* q    = alloc((size_t)N_NODES * D_HID);
  float* kbuf = alloc((size_t)N_NODES * D_HID);
  float* vbuf = alloc((size_t)N_NODES * D_HID);
  float* xr   = alloc((size_t)N_NODES * D_HID);
  float* agg  = alloc((size_t)N_NODES * D_HID);
  float* xcat = alloc((size_t)N_NODES * 256);
  float* ebuf = alloc((size_t)N_EDGES * D_HID);
  float* score = alloc((size_t)N_EDGES);
  float* mbuf  = alloc((size_t)N_NODES);
  float* denom = alloc((size_t)N_NODES);
  unsigned* mord = (unsigned*)alloc((size_t)N_NODES);

  const float inv_sqrt_d = 1.0f / sqrtf((float)D_HID);

  const dim3 gNodeTiles(N_NODES / 16);   // 3125
  const dim3 gEdgeTiles(N_EDGES / 16);   // 50000
  const dim3 b256(256);
  const dim3 gInit((N_NODES * D_HID + 255) / 256);
  const dim3 gNode1((N_NODES + 255) / 256);
  const dim3 gEdge1((N_EDGES + 255) / 256);
  const dim3 gEdgeWave(N_EDGES / 8);     // 100000 (8 waves/block)
  const dim3 gNodeWave(N_NODES / 8);     // 6250

  auto run_layer = [&](const float* X, int ldx, int K,
                       const float* Wq, const float* bq,
                       const float* Wk, const float* bk,
                       const float* Wv, const float* bv,
                       const float* We,
                       const float* Wskip, const float* bskip,
                       const float* Wbeta, int colOff) {
    // Dense projections (WMMA fp32 GEMMs, LDS-staged A panel)
    gemm_wmma_f32<<<gNodeTiles, b256, 0, stream>>>(X, ldx, Wq, bq, q, D_HID, K);
    gemm_wmma_f32<<<gNodeTiles, b256, 0, stream>>>(X, ldx, Wk, bk, kbuf, D_HID, K);
    gemm_wmma_f32<<<gNodeTiles, b256, 0, stream>>>(X, ldx, Wv, bv, vbuf, D_HID, K);
    gemm_wmma_f32<<<gNodeTiles, b256, 0, stream>>>(X, ldx, Wskip, bskip, xr, D_HID, K);
    // Edge embedding GEMM (no bias)
    gemm_wmma_f32<<<gEdgeTiles, b256, 0, stream>>>(eatt, 64, We, (const float*)nullptr,
                                                   ebuf, D_HID, 64);
    // Attention: segment softmax + weighted scatter-add
    init_attn<<<gInit, b256, 0, stream>>>(agg, mord, denom);
    edge_score<<<gEdgeWave, b256, 0, stream>>>(src, dst, q, kbuf, ebuf, score, mord,
                                               inv_sqrt_d);
    finalize_max<<<gNode1, b256, 0, stream>>>(mord, mbuf);
    edge_exp<<<gEdge1, b256, 0, stream>>>(dst, score, mbuf, denom);
    edge_aggregate<<<gEdgeWave, b256, 0, stream>>>(src, dst, score, denom, vbuf, ebuf,
                                                   agg);
    // Beta-gated skip + LeakyReLU, write into concat buffer
    beta_gate<<<gNodeWave, b256, 0, stream>>>(agg, xr, Wbeta, xcat, colOff);
  };

  // Layer 1: input x_in [N,256]
  run_layer(x_in, 256, 256, Wq1, bq1, Wk1, bk1, Wv1, bv1, We1, Wskip1, bskip1,
            Wbeta1, /*colOff=*/0);
  // Layer 2: input x1 = xcat[:, 0:128] (ld = 256)
  run_layer(xcat, 256, 128, Wq2, bq2, Wk2, bk2, Wv2, bv2, We2, Wskip2, bskip2,
            Wbeta2, /*colOff=*/128);
  // Output MLP: out[N,64] = xcat[N,256] @ Wm^T + bm  (4 column tiles -> 128 threads)
  gemm_wmma_f32<<<gNodeTiles, dim3(128), 0, stream>>>(xcat, 256, Wm, bm, outp, 64, 256);
}